// HierarchicalPDFSampler_74371653697772
// MI455X (gfx1250) — compile-verified
//
#include <hip/hip_runtime.h>

// Hierarchical PDF sampling (NeRF fine samples) for MI455X / gfx1250.
// One ray per wave32. Memory-bound: 1280 B/ray -> ~14.4 us floor @ 23.3 TB/s.
//
// Data path (gfx1250):
//   * double-buffered GLOBAL_LOAD_ASYNC_TO_LDS_B64 (ASYNCcnt) stages the next
//     ray's depth+weight rows while the current ray computes
//   * wave32 shfl scan builds the CDF, LDS holds cdf/mids/samples
//   * stable merge by ranking, staged in LDS, written back as 3 coalesced
//     non-temporal b64 stores per lane (full-cacheline write stream)

typedef __attribute__((ext_vector_type(2))) float f2;
typedef int v2i __attribute__((vector_size(2 * sizeof(int))));   // b64 payload type

#define NUM_FINE 128
#define NCOARSE  64
#define OUTW     (NCOARSE + NUM_FINE)   // 192
#define WPB      8                      // waves per block
#define BLOCK    (WPB * 32)

#if __has_builtin(__builtin_amdgcn_global_load_async_to_lds_b64) && \
    __has_builtin(__builtin_amdgcn_s_wait_asynccnt)
#define ASYNC_PIPE 1
#else
#define ASYNC_PIPE 0
#endif

#if ASYNC_PIPE
typedef __attribute__((address_space(1))) v2i* gptr_v2i;   // global (prints as __device__)
typedef __attribute__((address_space(3))) v2i* lptr_v2i;   // LDS    (prints as __shared__)

__device__ __forceinline__ void issue_row_async(const float* __restrict__ depth,
                                                const float* __restrict__ weights,
                                                size_t roff, f2* stage, int lane)
{
    // lane l copies 8B of the depth row into stage[0..31] and 8B of the
    // weight row into stage[32..63]; one wave instruction = one 256B row.
    __builtin_amdgcn_global_load_async_to_lds_b64(
        (gptr_v2i)(depth + roff + 2 * lane),
        (lptr_v2i)(stage + lane), 0, 0);
    __builtin_amdgcn_global_load_async_to_lds_b64(
        (gptr_v2i)(weights + roff + 2 * lane),
        (lptr_v2i)(stage + 32 + lane), 0, 0);
}
#endif

__global__ __launch_bounds__(BLOCK)
void hier_pdf_sampler_kernel(const float* __restrict__ depth,
                             const float* __restrict__ weights,
                             float* __restrict__ out,
                             int nrays)
{
    __shared__ f2    s_stage[WPB][2][64];      // [buf][0..31 depth f2 | 32..63 weight f2]
    __shared__ float s_cdf[WPB][64];           // 63 used (+1 pad)
    __shared__ float s_mid[WPB][64];           // 63 used (+1 pad)
    __shared__ float s_smp[WPB][NUM_FINE];
    __shared__ f2    s_out[WPB][OUTW / 2];     // merged row staging (192 floats)

    const int lane   = threadIdx.x & 31;
    const int wib    = threadIdx.x >> 5;
    const int gwave  = blockIdx.x * WPB + wib;
    const int nwaves = gridDim.x * WPB;

    float* cdfw = s_cdf[wib];
    float* midw = s_mid[wib];
    float* smpw = s_smp[wib];
    float* outw = (float*)s_out[wib];

    int cur = 0;
#if ASYNC_PIPE
    if (gwave < nrays)
        issue_row_async(depth, weights, (size_t)gwave * NCOARSE, s_stage[wib][0], lane);
#endif

    for (int r = gwave; r < nrays; r += nwaves) {
        const int rn = r + nwaves;

#if ASYNC_PIPE
        // Software pipeline: kick next ray's rows, then wait only for the
        // current buffer (async loads complete in order -> cnt<=2 is enough).
        if (rn < nrays) {
            issue_row_async(depth, weights, (size_t)rn * NCOARSE, s_stage[wib][cur ^ 1], lane);
            __builtin_amdgcn_s_wait_asynccnt(2);
        } else {
            __builtin_amdgcn_s_wait_asynccnt(0);
        }
        asm volatile("" ::: "memory");
        const f2 dp = s_stage[wib][cur][lane];        // depth[2l], depth[2l+1]
        const f2 wp = s_stage[wib][cur][32 + lane];   // weights[2l], weights[2l+1]
        const float* depw = (const float*)&s_stage[wib][cur][0];
#else
        // Fallback: coalesced b64 register loads + gfx1250 global_prefetch_b8.
        if (rn < nrays) {
            __builtin_prefetch(depth   + (size_t)rn * NCOARSE + lane * 2, 0, 0);
            __builtin_prefetch(weights + (size_t)rn * NCOARSE + lane * 2, 0, 0);
        }
        const f2 dp = ((const f2*)(depth   + (size_t)r * NCOARSE))[lane];
        const f2 wp = ((const f2*)(weights + (size_t)r * NCOARSE))[lane];
        s_stage[wib][0][lane] = dp;                   // stage depth row for the merge
        const float* depw = (const float*)&s_stage[wib][0][0];
#endif

        // ---- padded pdf source: v[j] = weights[j+1] + 1e-5 for j<62 else 0 ----
        float wnext = __shfl_down(wp.x, 1, 32);                            // weights[2l+2]
        float x0 = (2 * lane     < NCOARSE - 2) ? (wp.y  + 1e-5f) : 0.0f;  // v[2l]
        float x1 = (2 * lane + 1 < NCOARSE - 2) ? (wnext + 1e-5f) : 0.0f;  // v[2l+1]

        // ---- wave32 inclusive scan of per-lane pair sums (5 shfl steps) ----
        float s    = x0 + x1;
        float incl = s;
        #pragma unroll
        for (int d = 1; d < 32; d <<= 1) {
            float t = __shfl_up(incl, d, 32);
            if (lane >= d) incl += t;
        }
        float excl  = incl - s;
        float total = __shfl(incl, 31, 32);
        float inv   = 1.0f / total;

        // cdf[0] = 0; cdf[j] = cumsum(v)[j-1] / total
        if (lane == 0) cdfw[0] = 0.0f;
        cdfw[2 * lane + 1] = (excl + x0) * inv;          // j = 2l+1 (lane31 -> pad slot 63)
        if (lane < 31) cdfw[2 * lane + 2] = incl * inv;  // j = 2l+2, up to 62

        // mids staged to LDS for the inverse-CDF interpolation
        float dnext = __shfl_down(dp.x, 1, 32);          // depth[2l+2]
        midw[2 * lane]     = 0.5f * (dp.x + dp.y);       // mids[2l]
        midw[2 * lane + 1] = 0.5f * (dp.y + dnext);      // mids[2l+1] (slot 63 = pad)
        asm volatile("" ::: "memory");   // same-wave DS ops are in-order on CDNA5

        // ---- 4 fine samples per lane: inverse-CDF via binary search ----
        float sreg[4];
        #pragma unroll
        for (int j = 0; j < 4; ++j) {
            int   k = 4 * lane + j;
            float u = (float)k / (float)(NUM_FINE - 1);
            int lo = 0, hi = NCOARSE - 1;                // 63 cdf entries
            while (lo < hi) {                            // 6 iterations
                int mid = (lo + hi) >> 1;
                if (cdfw[mid] <= u) lo = mid + 1; else hi = mid;
            }
            int below = lo - 1; below = below < 0 ? 0 : below;
            int above = lo;     above = above > NCOARSE - 2 ? NCOARSE - 2 : above;
            float cb = cdfw[below], ca = cdfw[above];
            float bb = midw[below], ba = midw[above];
            float den = ca - cb;
            den = (den < 1e-5f) ? 1.0f : den;
            float t  = (u - cb) / den;
            float sv = bb + t * (ba - bb);
            smpw[k] = sv;
            sreg[j] = sv;
        }
        asm volatile("" ::: "memory");

        // ---- stable merge by ranking (both sequences sorted) into LDS ----
        // depth element i: position = i + |{samples <  d}|  (lower_bound, 7 steps)
        #pragma unroll
        for (int e = 0; e < 2; ++e) {
            int   i = 2 * lane + e;
            float d = e ? dp.y : dp.x;
            int lo = 0, hi = NUM_FINE;
            while (lo < hi) {
                int mid = (lo + hi) >> 1;
                if (smpw[mid] < d) lo = mid + 1; else hi = mid;
            }
            outw[i + lo] = d;
        }
        // sample element k: position = k + |{depth <= s}|  (upper_bound, 6 steps)
        #pragma unroll
        for (int j = 0; j < 4; ++j) {
            int   k  = 4 * lane + j;
            float sv = sreg[j];
            int lo = 0, hi = NCOARSE;
            while (lo < hi) {
                int mid = (lo + hi) >> 1;
                if (depw[mid] <= sv) lo = mid + 1; else hi = mid;
            }
            outw[k + lo] = sv;
        }
        asm volatile("" ::: "memory");

        // ---- coalesced non-temporal writeback: 3 full-line b64 stores/lane ----
        f2* orow2 = (f2*)(out + (size_t)r * OUTW);
        const f2* ow2 = s_out[wib];
        #pragma unroll
        for (int j = 0; j < 3; ++j) {
            int idx = lane + 32 * j;
            __builtin_nontemporal_store(ow2[idx], orow2 + idx);
        }
        asm volatile("" ::: "memory");   // LDS reused next iteration (in-order per wave)

        cur ^= 1;
    }
}

extern "C" void kernel_launch(void* const* d_in, const int* in_sizes, int n_in,
                              void* d_out, int out_size, void* d_ws, size_t ws_size,
                              hipStream_t stream)
{
    (void)n_in; (void)out_size; (void)d_ws; (void)ws_size;
    const float* depth   = (const float*)d_in[0];   // [R, 64] f32
    const float* weights = (const float*)d_in[1];   // [R, 64] f32
    float* out = (float*)d_out;                     // [R, 192] f32

    int nrays = in_sizes[0] / NCOARSE;              // 262144

    // 4096 blocks * 8 waves = 32768 waves -> 8 rays per wave: enough
    // parallelism to saturate HBM, enough loop depth for the async pipeline.
    int blocks = 4096;
    int max_blocks = (nrays + WPB - 1) / WPB;
    if (blocks > max_blocks) blocks = max_blocks;

    hipLaunchKernelGGL(hier_pdf_sampler_kernel, dim3(blocks), dim3(BLOCK), 0, stream,
                       depth, weights, out, nrays);
}